// AttExplainer_6528350290215
// MI455X (gfx1250) — compile-verified
//
#include <hip/hip_runtime.h>
#include <math.h>

#define EMB_DIM 128
#define KNEI    16
#define NGRAPHS 256
#define TILE    50

typedef __attribute__((ext_vector_type(2))) float v2f;
typedef __attribute__((ext_vector_type(8))) float v8f;

// ---------------- Kernel 0: zero the accumulator regions ----------------
__global__ void k_init(float* __restrict__ mean_pool, float* __restrict__ add_b,
                       float* __restrict__ add_inv_b, float* __restrict__ counts, int gd) {
    int i = blockIdx.x * blockDim.x + threadIdx.x;
    if (i < gd) { mean_pool[i] = 0.f; add_b[i] = 0.f; add_inv_b[i] = 0.f; }
    if (i < NGRAPHS) counts[i] = 0.f;
}

// ---------------- Kernel 1: gather + mean over K neighbors ----------------
// One block per motif, 128 threads across D: each gathered row is one
// coalesced 512B load. n_emb (102.4MB) fits in the 192MB L2, so the 819MB of
// gather traffic streams at L2 rate after first touch.
__global__ void k_embed(const float* __restrict__ n_emb, const int* __restrict__ midx,
                        float* __restrict__ embed, int M) {
    int m = blockIdx.x;
    if (m >= M) return;
    int d = threadIdx.x;
    __shared__ int sidx[KNEI];
    if (d < KNEI) sidx[d] = midx[m * KNEI + d];
    __syncthreads();
    float acc = 0.f;
#pragma unroll
    for (int k = 0; k < KNEI; ++k)
        acc += n_emb[(size_t)sidx[k] * EMB_DIM + d];
    embed[(size_t)m * EMB_DIM + d] = acc * (1.0f / 16.0f);
}

// ---------------- Kernel 2: logits via V_WMMA_F32_16X16X4_F32 ----------------
// Each wave computes logits for 16 motifs: D = A(16x4) * B(4x16) + C over
// K=128 in steps of 4. B broadcasts the att chunk across all 16 columns, so
// every column of D equals the dot product; lanes 0 and 16 (N==0) write M=0..7
// and M=8..15 respectively. Exact fp32 math (matches the fp32 reference).
__global__ void k_logits_wmma(const float* __restrict__ embed, const float* __restrict__ att,
                              float* __restrict__ logits, int Mtiles) {
    int wave = blockIdx.x * (blockDim.x >> 5) + (threadIdx.x >> 5);
    if (wave >= Mtiles) return;                 // wave-uniform exit: EXEC all-1s for WMMA
    int lane  = threadIdx.x & 31;
    int mbase = wave * 16;
    int mrow  = mbase + (lane & 15);
    int khalf = (lane >> 4) << 1;               // lanes 16-31 hold K+2,K+3
    const float* erow = embed + (size_t)mrow * EMB_DIM + khalf;
    const float* arow = att + khalf;
    v8f c = {};
#pragma unroll
    for (int kk = 0; kk < EMB_DIM; kk += 4) {
        v2f a = *(const v2f*)(erow + kk);       // A: M=lane%16, K=kk+khalf..+1
        v2f b = *(const v2f*)(arow + kk);       // B: K=kk+khalf..+1, same for all N
        c = __builtin_amdgcn_wmma_f32_16x16x4_f32(false, a, false, b,
                                                  (short)0, c, false, false);
    }
    if ((lane & 15) == 0) {                     // N==0 column holders
        int mb = mbase + ((lane >> 4) << 3);
#pragma unroll
        for (int j = 0; j < 8; ++j)
            logits[mb + j] = c[j] * 0.1f;       // fold the /10 temperature
    }
}

// Scalar tail for M % 16 motifs (not launched when M % 16 == 0).
__global__ void k_logits_rem(const float* __restrict__ embed, const float* __restrict__ att,
                             float* __restrict__ logits, int m0, int M) {
    int m = m0 + blockIdx.x * blockDim.x + threadIdx.x;
    if (m >= M) return;
    float s = 0.f;
    for (int k = 0; k < EMB_DIM; ++k) s += embed[(size_t)m * EMB_DIM + k] * att[k];
    logits[m] = s * 0.1f;
}

// ---------------- Kernel 3: softmax max + sum-of-exp (single block) ----------------
__global__ void k_softmax_reduce(const float* __restrict__ logits, float* __restrict__ scalars,
                                 int M) {
    __shared__ float red[1024];
    int tid = threadIdx.x;
    float mx = -INFINITY;
    for (int i = tid; i < M; i += 1024) mx = fmaxf(mx, logits[i]);
    red[tid] = mx; __syncthreads();
    for (int s = 512; s > 0; s >>= 1) {
        if (tid < s) red[tid] = fmaxf(red[tid], red[tid + s]);
        __syncthreads();
    }
    float gmax = red[0];
    __syncthreads();
    float sm = 0.f;
    for (int i = tid; i < M; i += 1024) sm += expf(logits[i] - gmax);
    red[tid] = sm; __syncthreads();
    for (int s = 512; s > 0; s >>= 1) {
        if (tid < s) red[tid] += red[tid + s];
        __syncthreads();
    }
    if (tid == 0) { scalars[0] = gmax; scalars[1] = red[0]; }
}

// ---------------- Kernel 4: gates + run-compacted segment sums ----------------
// batch is sorted, so a block-run of TILE consecutive motifs spans ~1 graph:
// accumulate in registers per run and flush with one atomicAdd per graph
// change (~50x fewer atomics than per-motif).
__global__ void k_scatter(const float* __restrict__ embed, const float* __restrict__ logits,
                          const int* __restrict__ batch, const float* __restrict__ scalars,
                          float* __restrict__ a_out, float* __restrict__ mean_acc,
                          float* __restrict__ add_b, float* __restrict__ add_inv_b,
                          float* __restrict__ counts, int M) {
    int d  = threadIdx.x;
    int m0 = blockIdx.x * TILE;
    if (m0 >= M) return;
    float gmax   = scalars[0];
    float invsum = 1.0f / scalars[1];
    int g_cur = -1;
    float accm = 0.f, accb = 0.f, acci = 0.f;
    int run = 0;
    for (int t = 0; t < TILE; ++t) {
        int m = m0 + t;
        if (m >= M) break;
        int g = batch[m];                      // block-uniform
        if (g != g_cur) {
            if (g_cur >= 0) {
                atomicAdd(&mean_acc[g_cur * EMB_DIM + d], accm);
                atomicAdd(&add_b[g_cur * EMB_DIM + d], accb);
                atomicAdd(&add_inv_b[g_cur * EMB_DIM + d], acci);
                if (d == 0) atomicAdd(&counts[g_cur], (float)run);
            }
            g_cur = g; accm = accb = acci = 0.f; run = 0;
        }
        float aval = expf(logits[m] - gmax) * invsum;
        float cg   = 1.0f / (1.0f + expf(-1000.0f * (aval - 0.1f)));
        float e    = embed[(size_t)m * EMB_DIM + d];
        accm += e * cg;
        accb += e * aval;
        acci += e * (1.0f - aval);
        ++run;
        if (d == 0) a_out[m] = aval;
    }
    if (g_cur >= 0) {
        atomicAdd(&mean_acc[g_cur * EMB_DIM + d], accm);
        atomicAdd(&add_b[g_cur * EMB_DIM + d], accb);
        atomicAdd(&add_inv_b[g_cur * EMB_DIM + d], acci);
        if (d == 0) atomicAdd(&counts[g_cur], (float)run);
    }
}

// ---------------- Kernel 5: mean_pool = seg_sum / max(counts, 1) ----------------
__global__ void k_finalize(float* __restrict__ mean_acc, const float* __restrict__ counts) {
    int i = blockIdx.x * blockDim.x + threadIdx.x;
    if (i < NGRAPHS * EMB_DIM)
        mean_acc[i] /= fmaxf(counts[i / EMB_DIM], 1.0f);
}

extern "C" void kernel_launch(void* const* d_in, const int* in_sizes, int n_in,
                              void* d_out, int out_size, void* d_ws, size_t ws_size,
                              hipStream_t stream) {
    const float* n_emb = (const float*)d_in[0];
    const float* att   = (const float*)d_in[1];
    const int*   midx  = (const int*)d_in[2];
    const int*   batch = (const int*)d_in[3];
    // d_in[4] = num_graphs (device scalar, fixed at 256 by the problem)

    const int M = in_sizes[3];                 // 100000 motifs

    // d_out layout: a[M] | embed[M*128] | mean_pool[256*128] | add_b | add_inv_b
    float* out       = (float*)d_out;
    float* a_out     = out;
    float* embed     = out + M;
    float* mean_pool = embed + (size_t)M * EMB_DIM;
    float* add_b     = mean_pool + NGRAPHS * EMB_DIM;
    float* add_inv_b = add_b + NGRAPHS * EMB_DIM;

    // workspace: logits[M] | scalars[2] | counts[256]
    float* ws      = (float*)d_ws;
    float* logits  = ws;
    float* scalars = ws + M;
    float* counts  = scalars + 2;

    const int gd = NGRAPHS * EMB_DIM;
    k_init<<<(gd + 255) / 256, 256, 0, stream>>>(mean_pool, add_b, add_inv_b, counts, gd);

    k_embed<<<M, EMB_DIM, 0, stream>>>(n_emb, midx, embed, M);

    int Mtiles = M / 16;
    if (Mtiles > 0) {
        int waves_per_block = 256 / 32;
        k_logits_wmma<<<(Mtiles + waves_per_block - 1) / waves_per_block, 256, 0, stream>>>(
            embed, att, logits, Mtiles);
    }
    int rem = M - Mtiles * 16;
    if (rem > 0)
        k_logits_rem<<<1, 64, 0, stream>>>(embed, att, logits, Mtiles * 16, M);

    k_softmax_reduce<<<1, 1024, 0, stream>>>(logits, scalars, M);

    k_scatter<<<(M + TILE - 1) / TILE, EMB_DIM, 0, stream>>>(
        embed, logits, batch, scalars, a_out, mean_pool, add_b, add_inv_b, counts, M);

    k_finalize<<<(gd + 255) / 256, 256, 0, stream>>>(mean_pool, counts);
}